// Model_64536178589887
// MI455X (gfx1250) — compile-verified
//
#include <hip/hip_runtime.h>

// Problem constants (match reference)
#define B_    4
#define CIN   3
#define H_    512
#define W_    512
#define KTAPS 9
#define COUT  3
#define HO_   510
#define WO_   510
#define KDIM  27                  // CIN * 3 * 3 (im2col K)
#define OC1   18                  // offset channels (2 * KTAPS)
#define NPIX  (B_ * HO_ * WO_)    // 1,040,400 output pixels (divisible by 16)

typedef __attribute__((ext_vector_type(2))) float v2f;
typedef __attribute__((ext_vector_type(8))) float v8f;

// im2col A-element: patch value for pixel (b,i,j), flattened k = c*9 + r*3 + s
__device__ __forceinline__ float loadA(const float* __restrict__ xb, int i, int j, int k) {
  if (k >= KDIM) return 0.0f;          // K padded 27 -> 28
  const int c = k / 9;
  const int r = (k - c * 9) / 3;
  const int s = k - c * 9 - r * 3;
  return xb[((size_t)c * H_ + (size_t)(i + r)) * W_ + (size_t)(j + s)];
}

// -----------------------------------------------------------------------------
// Kernel 1: offset-predicting conv as im2col GEMM via V_WMMA_F32_16X16X4_F32.
// One wave computes a 16-pixel x 18-channel tile of offs.
// A (16x4 fp32): lane L holds M = L&15, K = k0 + (L>>4)*2 .. +1 in {a.x, a.y}.
// B (4x16 fp32): lane L holds N = L&15, rows K mirrored like A halves.
// D (16x16 fp32, 8 VGPRs): vgpr v, lane L -> M = v + (L>=16 ? 8 : 0), N = L&15.
// Result stored pixel-major: offs[pixel][18]  (lanes 0..15 store contiguously).
// -----------------------------------------------------------------------------
__global__ __launch_bounds__(256) void offs_conv_wmma(
    const float* __restrict__ x, const float* __restrict__ w1,
    const float* __restrict__ b1, float* __restrict__ offs) {
  __shared__ float sw1[OC1 * KDIM];  // 486 floats
  __shared__ float sb1[OC1];
  for (int idx = threadIdx.x; idx < OC1 * KDIM; idx += 256) sw1[idx] = w1[idx];
  if (threadIdx.x < OC1) sb1[threadIdx.x] = b1[threadIdx.x];
  __syncthreads();

  const int lane = threadIdx.x & 31;
  const int tile = blockIdx.x * 8 + (threadIdx.x >> 5);  // one wave per tile
  if (tile >= NPIX / 16) return;                         // wave-uniform exit

  const int nloc  = lane & 15;        // column within tile / A-row pixel slot
  const int khalf = (lane >> 4) << 1; // 0 for lanes 0-15, 2 for lanes 16-31

  // A-row pixel for this lane
  const int m   = tile * 16 + nloc;
  const int b   = m / (HO_ * WO_);
  const int rem = m - b * (HO_ * WO_);
  const int i   = rem / WO_;
  const int j   = rem - i * WO_;
  const float* xb = x + (size_t)b * CIN * H_ * W_;

  v8f acc0 = {};  // output channels 0..15
  v8f acc1 = {};  // output channels 16..17 (rest zero-padded)

#pragma unroll
  for (int k0 = 0; k0 < KDIM; k0 += 4) {
    const int k = k0 + khalf;
    v2f a, bv0, bv1;
    a.x = loadA(xb, i, j, k);
    a.y = loadA(xb, i, j, k + 1);
    bv0.x = (k     < KDIM) ? sw1[nloc * KDIM + k]     : 0.0f;
    bv0.y = (k + 1 < KDIM) ? sw1[nloc * KDIM + k + 1] : 0.0f;
    const int n1 = nloc + 16;
    bv1.x = (n1 < OC1 && k     < KDIM) ? sw1[n1 * KDIM + k]     : 0.0f;
    bv1.y = (n1 < OC1 && k + 1 < KDIM) ? sw1[n1 * KDIM + k + 1] : 0.0f;
    acc0 = __builtin_amdgcn_wmma_f32_16x16x4_f32(
        false, a, false, bv0, (short)0, acc0, false, false);
    acc1 = __builtin_amdgcn_wmma_f32_16x16x4_f32(
        false, a, false, bv1, (short)0, acc1, false, false);
  }

  // Store D + bias, pixel-major [pixel][18]
  const int   mbase = tile * 16 + ((lane >> 4) << 3);
  const float bias0 = sb1[nloc];
  const float bias1 = (nloc < 2) ? sb1[16 + nloc] : 0.0f;
#pragma unroll
  for (int v = 0; v < 8; ++v) {
    float* dst = offs + (size_t)(mbase + v) * OC1;
    dst[nloc] = acc0[v] + bias0;
    if (nloc < 2) dst[16 + nloc] = acc1[v] + bias1;
  }
}

// -----------------------------------------------------------------------------
// Kernel 2: deformable sampling + 3x3x3 weighted reduction.
// One thread per output pixel; reads its 18 contiguous offsets, does 9 taps of
// bilinear gather (zero outside image, torchvision semantics), accumulates all
// 3 output channels.
// -----------------------------------------------------------------------------
__global__ __launch_bounds__(256) void deform_conv(
    const float* __restrict__ x, const float* __restrict__ offs,
    const float* __restrict__ w2, const float* __restrict__ b2,
    float* __restrict__ out) {
  __shared__ float sw2[COUT * KDIM];  // 81 floats
  __shared__ float sb2[COUT];
  if (threadIdx.x < COUT * KDIM) sw2[threadIdx.x] = w2[threadIdx.x];
  if (threadIdx.x < COUT) sb2[threadIdx.x] = b2[threadIdx.x];
  __syncthreads();

  const int p = blockIdx.x * 256 + threadIdx.x;
  if (p >= NPIX) return;
  const int b   = p / (HO_ * WO_);
  const int rem = p - b * (HO_ * WO_);
  const int i   = rem / WO_;
  const int j   = rem - i * WO_;
  const float* op = offs + (size_t)p * OC1;
  const float* xb = x + (size_t)b * CIN * H_ * W_;

  float a0 = sb2[0], a1 = sb2[1], a2 = sb2[2];

#pragma unroll
  for (int t = 0; t < KTAPS; ++t) {
    const int ky = t / 3;
    const int kx = t - ky * 3;
    const float sy = (float)(i + ky) + op[2 * t];
    const float sx = (float)(j + kx) + op[2 * t + 1];
    const float yf = floorf(sy), xf = floorf(sx);
    const float dy = sy - yf, dx = sx - xf;
    const int y0 = (int)yf, x0 = (int)xf;

    float v0 = 0.0f, v1 = 0.0f, v2 = 0.0f;
#pragma unroll
    for (int cy = 0; cy < 2; ++cy) {
      const int yy = y0 + cy;
      if (yy < 0 || yy >= H_) continue;
      const float wy = cy ? dy : (1.0f - dy);
#pragma unroll
      for (int cx = 0; cx < 2; ++cx) {
        const int xx = x0 + cx;
        if (xx < 0 || xx >= W_) continue;
        const float w = wy * (cx ? dx : (1.0f - dx));
        const size_t base = (size_t)yy * W_ + (size_t)xx;
        v0 += w * xb[base];
        v1 += w * xb[(size_t)H_ * W_ + base];
        v2 += w * xb[2 * (size_t)H_ * W_ + base];
      }
    }
    a0 += v0 * sw2[0 * KDIM + t] + v1 * sw2[0 * KDIM + 9 + t] + v2 * sw2[0 * KDIM + 18 + t];
    a1 += v0 * sw2[1 * KDIM + t] + v1 * sw2[1 * KDIM + 9 + t] + v2 * sw2[1 * KDIM + 18 + t];
    a2 += v0 * sw2[2 * KDIM + t] + v1 * sw2[2 * KDIM + 9 + t] + v2 * sw2[2 * KDIM + 18 + t];
  }

  const size_t hw = (size_t)HO_ * WO_;
  float* ob = out + (size_t)b * COUT * hw + (size_t)i * WO_ + (size_t)j;
  ob[0]      = a0;
  ob[hw]     = a1;
  ob[2 * hw] = a2;
}

extern "C" void kernel_launch(void* const* d_in, const int* in_sizes, int n_in,
                              void* d_out, int out_size, void* d_ws, size_t ws_size,
                              hipStream_t stream) {
  (void)in_sizes; (void)n_in; (void)out_size; (void)ws_size;
  const float* x  = (const float*)d_in[0];
  const float* w1 = (const float*)d_in[1];
  const float* b1 = (const float*)d_in[2];
  const float* w2 = (const float*)d_in[3];
  const float* b2 = (const float*)d_in[4];
  float* out  = (float*)d_out;
  float* offs = (float*)d_ws;  // NPIX * 18 floats = 74.9 MB scratch

  const int numTiles = NPIX / 16;            // 65,025 wave-tiles
  const int blocks1  = (numTiles + 7) / 8;   // 8 waves per 256-thread block
  offs_conv_wmma<<<blocks1, 256, 0, stream>>>(x, w1, b1, offs);

  const int blocks2 = (NPIX + 255) / 256;
  deform_conv<<<blocks2, 256, 0, stream>>>(x, offs, w2, b2, out);
}